// PoseGraph_42760694399316
// MI455X (gfx1250) — compile-verified
//
#include <hip/hip_runtime.h>
#include <math.h>

#define TILE 256
#define PSTR 20   // floats per pose slot in LDS: 16 data + 4 pad (80B, keeps b128 reads conflict-light)

struct V { float x, y, z; };
struct Q { float x, y, z, w; };

__device__ __forceinline__ V vcross(V a, V b) {
    V r; r.x = a.y*b.z - a.z*b.y; r.y = a.z*b.x - a.x*b.z; r.z = a.x*b.y - a.y*b.x; return r;
}
__device__ __forceinline__ V vadd(V a, V b)   { V r{a.x+b.x, a.y+b.y, a.z+b.z}; return r; }
__device__ __forceinline__ V vsub(V a, V b)   { V r{a.x-b.x, a.y-b.y, a.z-b.z}; return r; }
__device__ __forceinline__ V vscale(V a, float s) { V r{a.x*s, a.y*s, a.z*s}; return r; }
__device__ __forceinline__ float vdot(V a, V b) { return a.x*b.x + a.y*b.y + a.z*b.z; }

__device__ __forceinline__ Q qmul(Q a, Q b) {
    Q r;
    r.w = a.w*b.w - (a.x*b.x + a.y*b.y + a.z*b.z);
    V v1{a.x, a.y, a.z}, v2{b.x, b.y, b.z};
    V c = vcross(v1, v2);
    r.x = a.w*b.x + b.w*a.x + c.x;
    r.y = a.w*b.y + b.w*a.y + c.y;
    r.z = a.w*b.z + b.w*a.z + c.z;
    return r;
}
__device__ __forceinline__ V qrot(Q q, V v) {
    V u{q.x, q.y, q.z};
    V c1 = vcross(u, v);
    V t{c1.x + q.w*v.x, c1.y + q.w*v.y, c1.z + q.w*v.z};
    V c2 = vcross(u, t);
    V r{v.x + 2.0f*c2.x, v.y + 2.0f*c2.y, v.z + 2.0f*c2.z};
    return r;
}

__global__ __launch_bounds__(TILE) void pose_graph_err_kernel(
    const float* __restrict__ nodes,   // [N,7]  t(3) q(4)
    const float* __restrict__ poses,   // [E,4,4] row-major
    const int*   __restrict__ edges,   // [E,2]
    float* __restrict__ out,           // [E,6]
    int nE)
{
    __shared__ __align__(16) float shp[TILE * PSTR]; // staged pose tile (20KB)
    __shared__ float sho[TILE * 6];                  // output transpose tile (6KB)

    const int tid  = threadIdx.x;
    const int base = blockIdx.x * TILE;
    const int e    = base + tid;

    // ---- 1) fire async DMA: stream this block's 16KB pose tile into LDS ----
    // tile covers global float4 indices [base*4, base*4+1024); lanes contiguous.
    const long long maxf4 = (long long)nE * 4 - 1;
#pragma unroll
    for (int k = 0; k < 4; ++k) {
        int floc = k * TILE + tid;                 // 0..1023 local float4 index
        long long f = (long long)base * 4 + floc;  // global float4 index
        if (f > maxf4) f = maxf4;                  // clamp tail (harmless dup load)
        int eloc = floc >> 2, m = floc & 3;
        unsigned lds = (unsigned)(size_t)&shp[eloc * PSTR + m * 4]; // LDS byte offset (low 32 bits of flat addr)
        unsigned long long ga = (unsigned long long)(size_t)(poses + (size_t)f * 4);
        asm volatile("global_load_async_to_lds_b128 %0, %1, off"
                     :: "v"(lds), "v"(ga) : "memory");
    }

    // ---- 2) overlap: gather edge endpoints + node states (L2-resident) ----
    int ecl = (e < nE) ? e : (nE - 1);
    int i0 = edges[2 * ecl + 0];
    int i1 = edges[2 * ecl + 1];
    const float* p1 = nodes + 7ll * i0;
    const float* p2 = nodes + 7ll * i1;
    V t1{p1[0], p1[1], p1[2]}; Q q1{p1[3], p1[4], p1[5], p1[6]};
    V t2{p2[0], p2[1], p2[2]}; Q q2{p2[3], p2[4], p2[5], p2[6]};

    // relative pose n1^-1 * n2
    Q q1i{-q1.x, -q1.y, -q1.z, q1.w};
    V t12 = qrot(q1i, vsub(t2, t1));
    Q q12 = qmul(q1i, q2);

    // ---- 3) wait for tensor tile, read own pose from LDS ----
    asm volatile("s_wait_asynccnt 0" ::: "memory");
    __syncthreads();

    const float* M = &shp[tid * PSTR];
    float m00 = M[0], m01 = M[1], m02 = M[2],  tx = M[3];
    float m10 = M[4], m11 = M[5], m12 = M[6],  ty = M[7];
    float m20 = M[8], m21 = M[9], m22 = M[10], tz = M[11];

    // mat2SE3 (branchless copysign method, same clips as reference)
    float qw = 0.5f * sqrtf(fmaxf(1.0f + m00 + m11 + m22, 1e-12f));
    float sx = (m21 - m12 >= 0.0f) ? 1.0f : -1.0f;
    float sy = (m02 - m20 >= 0.0f) ? 1.0f : -1.0f;
    float sz = (m10 - m01 >= 0.0f) ? 1.0f : -1.0f;
    Q qp{ sx * 0.5f * sqrtf(fmaxf(1.0f + m00 - m11 - m22, 1e-12f)),
          sy * 0.5f * sqrtf(fmaxf(1.0f - m00 + m11 - m22, 1e-12f)),
          sz * 0.5f * sqrtf(fmaxf(1.0f - m00 - m11 + m22, 1e-12f)),
          qw };
    V tp{tx, ty, tz};

    // compose: err pose = pose * (n1^-1 n2)
    V te = vadd(tp, qrot(qp, t12));
    Q qe = qmul(qp, q12);

    // SO3 log
    float s  = (qe.w < 0.0f) ? -1.0f : 1.0f;
    V vq{ s * qe.x, s * qe.y, s * qe.z };
    float wq = s * qe.w;
    float n2 = vdot(vq, vq);
    float n  = sqrtf(fmaxf(n2, 1e-12f));
    float scale;
    if (n2 > 1e-8f) scale = 2.0f * atan2f(n, wq) / n;
    else            scale = 2.0f / wq - 2.0f * n2 / (3.0f * wq * wq * wq);
    V phi = vscale(vq, scale);

    // SE3 log translation part
    float th2  = vdot(phi, phi);
    float th   = sqrtf(fmaxf(th2, 1e-12f));
    float half = 0.5f * th;
    float coef;
    if (th2 > 1e-8f) {
        float sh = sinf(half);
        coef = (1.0f - half * cosf(half) / sh) / fmaxf(th2, 1e-12f);
    } else {
        coef = 1.0f / 12.0f + th2 / 720.0f;
    }
    V pxt = vcross(phi, te);
    V tau = vadd(vsub(te, vscale(pxt, 0.5f)), vscale(vcross(phi, pxt), coef));

    // weight: last edge gets KEY_EDGE_WEIGHT
    float wgt = (e == nE - 1) ? 0.1f : 1.0f;
    sho[tid * 6 + 0] = tau.x * wgt;
    sho[tid * 6 + 1] = tau.y * wgt;
    sho[tid * 6 + 2] = tau.z * wgt;
    sho[tid * 6 + 3] = phi.x * wgt;
    sho[tid * 6 + 4] = phi.y * wgt;
    sho[tid * 6 + 5] = phi.z * wgt;
    __syncthreads();

    // ---- 4) coalesced non-temporal streaming stores via LDS transpose ----
    const long long obase = (long long)base * 6;
    const long long total = (long long)nE * 6;
#pragma unroll
    for (int k = 0; k < 6; ++k) {
        long long idx = obase + k * TILE + tid;
        if (idx < total)
            __builtin_nontemporal_store(sho[k * TILE + tid], &out[idx]);
    }
}

extern "C" void kernel_launch(void* const* d_in, const int* in_sizes, int n_in,
                              void* d_out, int out_size, void* d_ws, size_t ws_size,
                              hipStream_t stream) {
    const float* nodes = (const float*)d_in[0];
    const float* poses = (const float*)d_in[1];
    const int*   edges = (const int*)d_in[2];
    float* out = (float*)d_out;

    int nE = in_sizes[2] / 2;   // edges is [E,2]
    int grid = (nE + TILE - 1) / TILE;
    pose_graph_err_kernel<<<dim3(grid), dim3(TILE), 0, stream>>>(nodes, poses, edges, out, nE);
}